// get_model_77309411328156
// MI455X (gfx1250) — compile-verified
//
#include <hip/hip_runtime.h>
#include <hip/hip_bf16.h>

// ---------------------------------------------------------------------------
// Point Transformer inference for MI455X (gfx1250, wave32).
// All linear layers run through a single WMMA-f32 GEMM kernel
// (V_WMMA_F32_16X16X4_F32, 32x32 output tile per wave, 2x2 blocking,
//  depth-2 software pipeline, strength-reduced pointer arithmetic).
// ---------------------------------------------------------------------------

typedef __attribute__((ext_vector_type(2))) float v2f;
typedef __attribute__((ext_vector_type(8))) float v8f;

#define FLAG_RELU 1
#define FLAG_AFF  2
#define FLAG_RES  4

#define WMMA4(A, B, C) __builtin_amdgcn_wmma_f32_16x16x4_f32( \
    false, (A), false, (B), (short)0, (C), false, false)

// Y[m][n] = act( (X@W + bias) * g + bt ) + res  (flag-controlled epilogue)
// X:[M,K] row-major, W:[K,N] row-major.
__global__ void gemm_kernel(const float* __restrict__ X, const float* __restrict__ W,
                            const float* __restrict__ bias,
                            const float* __restrict__ g, const float* __restrict__ bt,
                            const float* __restrict__ res, float* __restrict__ Y,
                            int M, int K, int N, int flags)
{
    int lane = threadIdx.x & 31;
    int wv   = threadIdx.x >> 5;
    int tm   = (blockIdx.y * 4 + wv) * 32;
    int tn   = blockIdx.x * 32;
    if (tm >= M) return;                 // wave-uniform
    int half = lane >> 4;                // 0: lanes 0-15, 1: lanes 16-31
    int q    = lane & 15;

    // Clamped addresses: garbage from clamped rows/cols only reaches
    // accumulator entries whose (row,col) is out of range and never stored.
    int r0 = tm + q;       if (r0 > M - 1) r0 = M - 1;
    int r1 = tm + 16 + q;  if (r1 > M - 1) r1 = M - 1;
    int c0 = tn + q;       if (c0 > N - 1) c0 = N - 1;
    int c1 = tn + 16 + q;  if (c1 > N - 1) c1 = N - 1;
    const float* Xr0 = X + (size_t)r0 * K;
    const float* Xr1 = X + (size_t)r1 * K;
    const float* Wc0 = W + c0;
    const float* Wc1 = W + c1;

    v8f z = {0.f,0.f,0.f,0.f,0.f,0.f,0.f,0.f};
    v8f acc00 = z, acc01 = z, acc10 = z, acc11 = z;
    const int koff = half * 2;           // lanes 16-31 hold K=2,3 of each chunk
    const int K4 = K & ~3;

    // Strength-reduced streaming pointers (advance by one K-chunk at a time).
    const float* pA0 = Xr0 + koff;                    // A row r0, k = koff
    const float* pA1 = Xr1 + koff;                    // A row r1
    const float* pB00 = Wc0 + (size_t)koff * N;       // B col c0, row ka
    const float* pB01 = pB00 + N;                     // B col c0, row ka+1
    const float* pB10 = Wc1 + (size_t)koff * N;       // B col c1, row ka
    const float* pB11 = pB10 + N;                     // B col c1, row ka+1
    const size_t bstep = (size_t)4 * N;

    // load one K-chunk into a register set and advance the stream pointers
#define LOADSET(SA0, SA1, SB0, SB1)                                          \
    do {                                                                     \
        SA0.x = pA0[0]; SA0.y = pA0[1];                                      \
        SA1.x = pA1[0]; SA1.y = pA1[1];                                      \
        SB0.x = pB00[0]; SB0.y = pB01[0];                                    \
        SB1.x = pB10[0]; SB1.y = pB11[0];                                    \
        pA0 += 4; pA1 += 4;                                                  \
        pB00 += bstep; pB01 += bstep; pB10 += bstep; pB11 += bstep;          \
    } while (0)

#define MMASET(SA0, SA1, SB0, SB1)                                           \
    do {                                                                     \
        acc00 = WMMA4(SA0, SB0, acc00);                                      \
        acc01 = WMMA4(SA0, SB1, acc01);                                      \
        acc10 = WMMA4(SA1, SB0, acc10);                                      \
        acc11 = WMMA4(SA1, SB1, acc11);                                      \
    } while (0)

    if (K4 >= 8) {                       // depth-2 software pipeline
        v2f a0A, a1A, b0A, b1A, a0B, a1B, b0B, b1B;
        LOADSET(a0A, a1A, b0A, b1A);     // chunk 0
        LOADSET(a0B, a1B, b0B, b1B);     // chunk 1
        int kk = 8;
        for (; kk + 8 <= K4; kk += 8) {
            MMASET(a0A, a1A, b0A, b1A);
            LOADSET(a0A, a1A, b0A, b1A);
            MMASET(a0B, a1B, b0B, b1B);
            LOADSET(a0B, a1B, b0B, b1B);
        }
        if (kk < K4) {                   // one odd chunk remains (K4 % 8 == 4)
            MMASET(a0A, a1A, b0A, b1A);
            LOADSET(a0A, a1A, b0A, b1A);
            MMASET(a0B, a1B, b0B, b1B);
            MMASET(a0A, a1A, b0A, b1A);
        } else {
            MMASET(a0A, a1A, b0A, b1A);
            MMASET(a0B, a1B, b0B, b1B);
        }
    } else if (K4 == 4) {
        v2f a0A, a1A, b0A, b1A;
        LOADSET(a0A, a1A, b0A, b1A);
        MMASET(a0A, a1A, b0A, b1A);
    }
#undef LOADSET

    if (K4 < K) {                        // K tail: branchless select-to-zero on A
        int ka  = K4 + koff;
        int k0c = (ka     < K) ? ka     : K - 1;
        int k1c = (ka + 1 < K) ? ka + 1 : K - 1;
        float m0 = (ka     < K) ? 1.f : 0.f;
        float m1 = (ka + 1 < K) ? 1.f : 0.f;
        v2f a0, a1, b0, b1;
        a0.x = Xr0[k0c] * m0;  a0.y = Xr0[k1c] * m1;
        a1.x = Xr1[k0c] * m0;  a1.y = Xr1[k1c] * m1;
        b0.x = Wc0[(size_t)k0c * N];  b0.y = Wc0[(size_t)k1c * N];
        b1.x = Wc1[(size_t)k0c * N];  b1.y = Wc1[(size_t)k1c * N];
        MMASET(a0, a1, b0, b1);
    }
#undef MMASET

    // Epilogue. C/D layout: VGPR r -> row (base + r) lanes 0-15, (base + r + 8)
    // lanes 16-31; column = lane&15 within each 16-wide tile.
#define EPILOG(ACC, RBASE, COL)                                              \
    do {                                                                     \
        int col_ = (COL);                                                    \
        if (col_ < N) {                                                      \
            float bb = bias ? bias[col_] : 0.f;                              \
            float gg = (flags & FLAG_AFF) ? g[col_]  : 1.f;                  \
            float tt = (flags & FLAG_AFF) ? bt[col_] : 0.f;                  \
            _Pragma("unroll")                                                \
            for (int r = 0; r < 8; ++r) {                                    \
                int row = (RBASE) + r + half * 8;                            \
                if (row < M) {                                               \
                    float v = ACC[r] + bb;                                   \
                    if (flags & FLAG_AFF)  v = v * gg + tt;                  \
                    if (flags & FLAG_RELU) v = fmaxf(v, 0.f);                \
                    size_t o = (size_t)row * N + col_;                       \
                    if (flags & FLAG_RES)  v += res[o];                      \
                    Y[o] = v;                                                \
                }                                                            \
            }                                                                \
        }                                                                    \
    } while (0)

    EPILOG(acc00, tm,      tn + q);
    EPILOG(acc01, tm,      tn + 16 + q);
    EPILOG(acc10, tm + 16, tn + q);
    EPILOG(acc11, tm + 16, tn + 16 + q);
#undef EPILOG
}

// ---------------------------------------------------------------------------
// KNN: thread-per-query, insertion list of k<=64 (matches top_k(-d) stability:
// strict compares keep the earliest index on ties).
// ---------------------------------------------------------------------------
#define KMAX 64
__global__ void knn_kernel(const float* __restrict__ ref, const float* __restrict__ query,
                           int* __restrict__ idx, int Bb, int Nref, int M, int k)
{
    int t = blockIdx.x * blockDim.x + threadIdx.x;
    if (t >= Bb * M) return;
    int b = t / M, m = t - b * M;
    const float* qp = query + ((size_t)b * M + m) * 3;
    float qx = qp[0], qy = qp[1], qz = qp[2];
    const float* rp = ref + (size_t)b * Nref * 3;

    float bd[KMAX]; int bi[KMAX];
    for (int i = 0; i < k; ++i) { bd[i] = 3.4e38f; bi[i] = 0; }
    for (int j = 0; j < Nref; ++j) {
        float dx = qx - rp[j*3], dy = qy - rp[j*3+1], dz = qz - rp[j*3+2];
        float d = dx*dx + dy*dy + dz*dz;
        if (d < bd[k-1]) {
            int p = k - 1;
            while (p > 0 && bd[p-1] > d) { bd[p] = bd[p-1]; bi[p] = bi[p-1]; --p; }
            bd[p] = d; bi[p] = j;
        }
    }
    int* op = idx + ((size_t)b * M + m) * k;
    for (int i = 0; i < k; ++i) op[i] = bi[i];
}

// ---------------------------------------------------------------------------
// Farthest point sampling: one block per batch, LDS min-distance array,
// argmax tree reduction with first-occurrence tie-break (jnp.argmax).
// ---------------------------------------------------------------------------
__global__ void fps_kernel(const float* __restrict__ xyz, int n, int npoint,
                           int* __restrict__ out)
{
    int b = blockIdx.x, tid = threadIdx.x;
    __shared__ float mind[4096];
    __shared__ float rv[256];
    __shared__ int   ri[256];
    __shared__ int   s_last;
    const float* xp = xyz + (size_t)b * n * 3;

    for (int i = tid; i < n; i += 256) mind[i] = 1e10f;
    if (tid == 0) { out[(size_t)b * npoint] = 0; s_last = 0; }
    __syncthreads();

    for (int s = 1; s < npoint; ++s) {
        int last = s_last;
        float px = xp[last*3], py = xp[last*3+1], pz = xp[last*3+2];
        float bv = -1.f; int bidx = 0x7fffffff;
        for (int i = tid; i < n; i += 256) {
            float dx = xp[i*3]-px, dy = xp[i*3+1]-py, dz = xp[i*3+2]-pz;
            float d  = dx*dx + dy*dy + dz*dz;
            float md = fminf(mind[i], d);
            mind[i] = md;
            if (md > bv) { bv = md; bidx = i; }   // strict: first max within thread
        }
        rv[tid] = bv; ri[tid] = bidx;
        __syncthreads();
        for (int off = 128; off > 0; off >>= 1) {
            if (tid < off) {
                float ov = rv[tid + off]; int oi = ri[tid + off];
                if (ov > rv[tid] || (ov == rv[tid] && oi < ri[tid])) {
                    rv[tid] = ov; ri[tid] = oi;
                }
            }
            __syncthreads();
        }
        if (tid == 0) { s_last = ri[0]; out[(size_t)b * npoint + s] = ri[0]; }
        __syncthreads();
    }
}

// out[b,mk,:D] = src[b, idx[b*Mk+mk], :D]
__global__ void gather_kernel(const float* __restrict__ src, const int* __restrict__ idx,
                              float* __restrict__ out, int Bb, int n, int D, int Mk)
{
    size_t t = (size_t)blockIdx.x * blockDim.x + threadIdx.x;
    size_t total = (size_t)Bb * Mk * D;
    if (t >= total) return;
    int d = (int)(t % D);
    size_t r = t / D;
    int b = (int)(r / Mk);
    int j = idx[r];
    out[t] = src[((size_t)b * n + j) * D + d];
}

// rel[b,m,kk,c] = xyz[b,m,c] - xyz[b, idx[b,m,kk], c]
__global__ void ptrel_kernel(const float* __restrict__ xyz, const int* __restrict__ idx,
                             float* __restrict__ rel, int Bb, int n, int k)
{
    size_t t = (size_t)blockIdx.x * blockDim.x + threadIdx.x;
    size_t total = (size_t)Bb * n * k * 3;
    if (t >= total) return;
    int c  = (int)(t % 3);
    size_t r = t / 3;                       // b*n*k + m*k + kk
    size_t bm = r / k;                      // b*n + m
    int b = (int)(bm / n);
    int j = idx[r];
    rel[t] = xyz[bm * 3 + c] - xyz[((size_t)b * n + j) * 3 + c];
}

// out[b,m,kk, 0:3]   = xyz[b,j] - new_xyz[b,m]
// out[b,m,kk, 3:3+C] = feat[b,j]
__global__ void tdgroup_kernel(const float* __restrict__ xyz, const float* __restrict__ feat,
                               const float* __restrict__ nxyz, const int* __restrict__ idx,
                               float* __restrict__ out, int Bb, int n, int C, int M, int k)
{
    int D = 3 + C;
    size_t t = (size_t)blockIdx.x * blockDim.x + threadIdx.x;
    size_t total = (size_t)Bb * M * k * D;
    if (t >= total) return;
    int d = (int)(t % D);
    size_t r = t / D;                        // (b*M+m)*k + kk
    size_t bm = r / k;                       // b*M + m
    int b = (int)(bm / M);
    int j = idx[r];
    if (d < 3) out[t] = xyz[((size_t)b * n + j) * 3 + d] - nxyz[bm * 3 + d];
    else       out[t] = feat[((size_t)b * n + j) * C + (d - 3)];
}

// t[bm,kk,d] = q[bm,d] - kf[bm,kk,d] + pos[bm,kk,d]
__global__ void qkp_kernel(const float* __restrict__ q, const float* __restrict__ kf,
                           const float* __restrict__ pos, float* __restrict__ t,
                           int BM, int k, int D)
{
    size_t i = (size_t)blockIdx.x * blockDim.x + threadIdx.x;
    size_t total = (size_t)BM * k * D;
    if (i >= total) return;
    int d = (int)(i % D);
    size_t bm = (i / D) / k;
    t[i] = q[bm * D + d] - kf[i] + pos[i];
}

// softmax over the k axis of a[BM,k,D], per (bm,d)
__global__ void softmaxk_kernel(float* __restrict__ a, int BM, int k, int D)
{
    int t = blockIdx.x * blockDim.x + threadIdx.x;
    if (t >= BM * D) return;
    int d = t % D, bm = t / D;
    float* p = a + (size_t)bm * k * D + d;
    float mx = -3.4e38f;
    for (int kk = 0; kk < k; ++kk) mx = fmaxf(mx, p[(size_t)kk * D]);
    float sum = 0.f;
    for (int kk = 0; kk < k; ++kk) {
        float e = __expf(p[(size_t)kk * D] - mx);
        p[(size_t)kk * D] = e; sum += e;
    }
    float inv = 1.f / sum;
    for (int kk = 0; kk < k; ++kk) p[(size_t)kk * D] *= inv;
}

// out[bm,d] = sum_k attn[bm,kk,d] * (v[bm,kk,d] + pos[bm,kk,d])
__global__ void attnsum_kernel(const float* __restrict__ attn, const float* __restrict__ v,
                               const float* __restrict__ pos, float* __restrict__ out,
                               int BM, int k, int D)
{
    int t = blockIdx.x * blockDim.x + threadIdx.x;
    if (t >= BM * D) return;
    int d = t % D, bm = t / D;
    size_t base = (size_t)bm * k * D + d;
    float s = 0.f;
    for (int kk = 0; kk < k; ++kk) {
        size_t o = base + (size_t)kk * D;
        s += attn[o] * (v[o] + pos[o]);
    }
    out[(size_t)bm * D + d] = s;
}

// out[bm,d] = max_k h[bm,kk,d]
__global__ void maxpoolk_kernel(const float* __restrict__ h, float* __restrict__ out,
                                int BM, int k, int D)
{
    int t = blockIdx.x * blockDim.x + threadIdx.x;
    if (t >= BM * D) return;
    int d = t % D, bm = t / D;
    size_t base = (size_t)bm * k * D + d;
    float m = h[base];
    for (int kk = 1; kk < k; ++kk) m = fmaxf(m, h[base + (size_t)kk * D]);
    out[(size_t)bm * D + d] = m;
}

// out[b,d] = mean_n f[b,n,d]
__global__ void meanpool_kernel(const float* __restrict__ f, float* __restrict__ out,
                                int Bb, int n, int D)
{
    int t = blockIdx.x * blockDim.x + threadIdx.x;
    if (t >= Bb * D) return;
    int d = t % D, b = t / D;
    float s = 0.f;
    for (int i = 0; i < n; ++i) s += f[((size_t)b * n + i) * D + d];
    out[(size_t)b * D + d] = s / (float)n;
}

// split inputs [B,6,N] -> xyz [B,N,3], f6 [B,N,6]
__global__ void split_kernel(const float* __restrict__ in, float* __restrict__ xyz,
                             float* __restrict__ f6, int Bb, int n)
{
    int t = blockIdx.x * blockDim.x + threadIdx.x;
    if (t >= Bb * n) return;
    int m = t % n, b = t / n;
    for (int c = 0; c < 6; ++c) {
        float v = in[((size_t)b * 6 + c) * n + m];
        f6[(size_t)t * 6 + c] = v;
        if (c < 3) xyz[(size_t)t * 3 + c] = v;
    }
}

__global__ void logsoftmax_kernel(const float* __restrict__ x, float* __restrict__ out,
                                  int Bb, int C)
{
    int b = blockIdx.x * blockDim.x + threadIdx.x;
    if (b >= Bb) return;
    const float* p = x + (size_t)b * C;
    float mx = p[0];
    for (int c = 1; c < C; ++c) mx = fmaxf(mx, p[c]);
    float s = 0.f;
    for (int c = 0; c < C; ++c) s += __expf(p[c] - mx);
    float lse = mx + __logf(s);
    for (int c = 0; c < C; ++c) out[(size_t)b * C + c] = p[c] - lse;
}

// ---------------------------------------------------------------------------
// Host orchestration
// ---------------------------------------------------------------------------
extern "C" void kernel_launch(void* const* d_in, const int* in_sizes, int n_in,
                              void* d_out, int out_size, void* d_ws, size_t ws_size,
                              hipStream_t stream)
{
    (void)in_sizes; (void)n_in; (void)out_size; (void)ws_size;
    const int Bc = 8, N0 = 4096;
    const int ns[5] = {4096, 1024, 256, 64, 16};
    const int ds[5] = {32, 64, 128, 256, 512};
    const int ks[5] = {16, 32, 64, 16, 16};
    const int ptIdx[5] = {9, 31, 53, 75, 97};
    const int tdIdx[4] = {27, 49, 71, 93};

    const float* inp = (const float*)d_in[0];
    auto P = [&](int i) { return (const float*)d_in[i]; };

    // bump allocator over d_ws (peak ~290 MB: 4x64MB attention-sized regions)
    char* base = (char*)d_ws;
    size_t off = 0;
    auto alloc = [&](size_t bytes) -> void* {
        void* p = base + off;
        off += (bytes + 255) & ~(size_t)255;
        return p;
    };

    const size_t ASZ = 16777216;                    // max B*n*k*d elements
    float* A1 = (float*)alloc(ASZ * 4);
    float* A2 = (float*)alloc(ASZ * 4);
    float* A3 = (float*)alloc(ASZ * 4);
    float* A4 = (float*)alloc(ASZ * 4);
    float* xbuf  = (float*)alloc((size_t)Bc * N0 * 32 * 4);   // max B*n*d
    float* qbuf  = (float*)alloc((size_t)Bc * N0 * 32 * 4);
    float* t0buf = (float*)alloc((size_t)Bc * N0 * 32 * 4);
    float* relb  = (float*)alloc((size_t)Bc * N0 * 16 * 3 * 4);
    int*   idxb  = (int*)alloc((size_t)Bc * N0 * 16 * 4);     // max B*n*k indices
    int*   fidxb = (int*)alloc((size_t)Bc * 1024 * 4);
    float* f6    = (float*)alloc((size_t)Bc * N0 * 6 * 4);
    float* xyzL[5]; float* featL[5];
    for (int l = 0; l < 5; ++l) {
        xyzL[l]  = (float*)alloc((size_t)Bc * ns[l] * 3 * 4);
        featL[l] = (float*)alloc((size_t)Bc * ns[l] * ds[l] * 4);
    }
    float* pooled = (float*)alloc((size_t)Bc * 512 * 4);
    float* fc1o   = (float*)alloc((size_t)Bc * 256 * 4);
    float* logits = (float*)alloc((size_t)Bc * 40 * 4);

    auto run_gemm = [&](const float* X, const float* W, const float* b,
                        const float* g, const float* bt, const float* res,
                        float* Y, int M, int K, int N, int flags) {
        dim3 blk(128, 1, 1);
        dim3 grd((N + 31) / 32, (M + 127) / 128, 1);
        gemm_kernel<<<grd, blk, 0, stream>>>(X, W, b, g, bt, res, Y, M, K, N, flags);
    };
    auto g1 = [](size_t total) { return dim3((unsigned)((total + 255) / 256), 1, 1); };

    // pt block: feat updated in place
    auto pt_block = [&](const float* xyz, float* feat, int n, int d, int k, int pi) {
        int BM = Bc * n;
        size_t BMK = (size_t)BM * k;
        knn_kernel<<<g1(BM), 256, 0, stream>>>(xyz, xyz, idxb, Bc, n, n, k);
        run_gemm(feat, P(pi+0),  P(pi+1),  0, 0, 0, xbuf,  BM, d, d, 0);          // top
        run_gemm(xbuf, P(pi+2),  P(pi+3),  0, 0, 0, qbuf,  BM, d, d, 0);          // phi
        run_gemm(xbuf, P(pi+4),  P(pi+5),  0, 0, 0, t0buf, BM, d, d, 0);          // psi
        gather_kernel<<<g1(BMK * d), 256, 0, stream>>>(t0buf, idxb, A1, Bc, n, d, n * k);
        run_gemm(xbuf, P(pi+6),  P(pi+7),  0, 0, 0, t0buf, BM, d, d, 0);          // alpha
        gather_kernel<<<g1(BMK * d), 256, 0, stream>>>(t0buf, idxb, A2, Bc, n, d, n * k);
        ptrel_kernel<<<g1(BMK * 3), 256, 0, stream>>>(xyz, idxb, relb, Bc, n, k);
        run_gemm(relb, P(pi+12), P(pi+13), 0, 0, 0, A3, (int)BMK, 3, d, FLAG_RELU); // relu(d1)
        run_gemm(A3,   P(pi+14), P(pi+15), 0, 0, 0, A4, (int)BMK, d, d, 0);         // pos = d2
        qkp_kernel<<<g1(BMK * d), 256, 0, stream>>>(qbuf, A1, A4, A3, BM, k, d);    // t
        run_gemm(A3,   P(pi+8),  P(pi+9),  0, 0, 0, A1, (int)BMK, d, d, FLAG_RELU); // relu(g1)
        run_gemm(A1,   P(pi+10), P(pi+11), 0, 0, 0, A3, (int)BMK, d, d, 0);         // a = g2
        softmaxk_kernel<<<g1((size_t)BM * d), 256, 0, stream>>>(A3, BM, k, d);
        attnsum_kernel<<<g1((size_t)BM * d), 256, 0, stream>>>(A3, A2, A4, xbuf, BM, k, d);
        run_gemm(xbuf, P(pi+16), P(pi+17), 0, 0, feat, feat, BM, d, d, FLAG_RES);   // down + res
    };

    auto td_layer = [&](const float* xyz, const float* feat, int n, int Cin,
                        float* nxyz, float* nfeat, int npoint, int Cout, int k, int ti) {
        fps_kernel<<<Bc, 256, 0, stream>>>(xyz, n, npoint, fidxb);
        gather_kernel<<<g1((size_t)Bc * npoint * 3), 256, 0, stream>>>(
            xyz, fidxb, nxyz, Bc, n, 3, npoint);
        knn_kernel<<<g1(Bc * npoint), 256, 0, stream>>>(xyz, nxyz, idxb, Bc, n, npoint, k);
        int Din = Cin + 3;
        size_t rows = (size_t)Bc * npoint * k;
        tdgroup_kernel<<<g1(rows * Din), 256, 0, stream>>>(
            xyz, feat, nxyz, idxb, A1, Bc, n, Cin, npoint, k);
        run_gemm(A1, P(ti+0), P(ti+1), P(ti+2), P(ti+3), 0, A2,
                 (int)rows, Din, Cout, FLAG_RELU | FLAG_AFF);
        maxpoolk_kernel<<<g1((size_t)Bc * npoint * Cout), 256, 0, stream>>>(
            A2, nfeat, Bc * npoint, k, Cout);
    };

    // --- forward pass -----------------------------------------------------
    split_kernel<<<g1(Bc * N0), 256, 0, stream>>>(inp, xyzL[0], f6, Bc, N0);
    run_gemm(f6,   P(1), P(2), P(3), P(4), 0, xbuf,     Bc * N0, 6,  32, FLAG_RELU | FLAG_AFF);
    run_gemm(xbuf, P(5), P(6), P(7), P(8), 0, featL[0], Bc * N0, 32, 32, FLAG_AFF);

    pt_block(xyzL[0], featL[0], ns[0], ds[0], ks[0], ptIdx[0]);
    for (int l = 0; l < 4; ++l) {
        td_layer(xyzL[l], featL[l], ns[l], ds[l],
                 xyzL[l+1], featL[l+1], ns[l+1], ds[l+1], ks[l+1], tdIdx[l]);
        pt_block(xyzL[l+1], featL[l+1], ns[l+1], ds[l+1], ks[l+1], ptIdx[l+1]);
    }

    meanpool_kernel<<<g1(Bc * 512), 256, 0, stream>>>(featL[4], pooled, Bc, 16, 512);
    run_gemm(pooled, P(115), P(116), P(117), P(118), 0, fc1o, Bc, 512, 256,
             FLAG_RELU | FLAG_AFF);
    run_gemm(fc1o, P(119), P(120), 0, 0, 0, logits, Bc, 256, 40, 0);
    logsoftmax_kernel<<<1, 32, 0, stream>>>(logits, (float*)d_out, Bc, 40);
}